// GAT_5119601017050
// MI455X (gfx1250) — compile-verified
//
#include <hip/hip_runtime.h>
#include <hip/hip_bf16.h>

#define N_NODES 50000
#define N_EDGES 800000
#define IN_DIM  256
#define HEADS   8
#define HID     64
#define WIDTH   512
#define NEG_SLOPE 0.2f
#define SM_EPS 1e-16f
#define LN_EPS 1e-5f

typedef float v2f __attribute__((ext_vector_type(2)));
typedef float v8f __attribute__((ext_vector_type(8)));
typedef int   v4i __attribute__((ext_vector_type(4)));

// ---- async global->LDS (CDNA5 ASYNCcnt path) with safe fallback -------------
#if defined(__has_builtin)
#if __has_builtin(__builtin_amdgcn_global_load_async_to_lds_b128)
#define HAVE_ASYNC_LDS 1
#endif
#endif

__device__ __forceinline__ void async_copy_b128(float* lds_dst, const float* gsrc) {
#ifdef HAVE_ASYNC_LDS
    // signature (probe-confirmed): (v4i addrspace(1)*, v4i addrspace(3)*, Ii offset, Ii cpol)
    __builtin_amdgcn_global_load_async_to_lds_b128(
        (__attribute__((address_space(1))) v4i*)gsrc,
        (__attribute__((address_space(3))) v4i*)lds_dst,
        0, 0);
#else
    *(float4*)lds_dst = *(const float4*)gsrc;
#endif
}
__device__ __forceinline__ void async_wait_all() {
#ifdef HAVE_ASYNC_LDS
#if __has_builtin(__builtin_amdgcn_s_wait_asynccnt)
    __builtin_amdgcn_s_wait_asynccnt(0);
#else
    asm volatile("s_wait_asynccnt 0" ::: "memory");
#endif
#endif
}

// ---- monotonic float <-> uint encoding so atomicMax(uint) orders like float max
__device__ __forceinline__ unsigned encF(float f) {
    unsigned u = __float_as_uint(f);
    return (u & 0x80000000u) ? ~u : (u | 0x80000000u);
}
__device__ __forceinline__ float decF(unsigned u) {
    return (u & 0x80000000u) ? __uint_as_float(u & 0x7FFFFFFFu)
                             : __uint_as_float(~u);
}
#define ENC_NEG_INF 0x007FFFFFu   // encF(-inf)

// ===================== WMMA fp32 GEMM: H[N,512] = X[N,K] @ W[K,512] ==========
// Block = 8 waves computes one 16-row x 512-col panel (grid = N/16 = 3125).
// Each wave owns 4 16x16 N-tiles (cols wave*64..wave*64+63), 4 v8f accumulators.
// Per K-chunk of 16: stage A(16x16, stride 18) + B(16x512, stride 520) in LDS
// (async global->LDS), then 4 k4-steps x 4 tiles = 16 WMMAs, reusing one A
// fragment per k4-step across the 4 tiles.
__global__ void __launch_bounds__(256) gemm_wmma(const float* __restrict__ X,
                                                 const float* __restrict__ W,
                                                 float* __restrict__ H,
                                                 int K) {
    __shared__ float Asm[16 * 18];    // padded: conflict-free b64 fragment reads
    __shared__ float Bsm[16 * 520];   // padded: lane-halves hit disjoint banks

    const int tid  = threadIdx.x;
    const int lane = tid & 31;
    const int wave = tid >> 5;
    const int tm   = blockIdx.x;
    const int r    = lane & 15;
    const int kk   = (lane >> 4) << 1;      // 0 or 2 (K split across lane halves)
    const int wcol = wave * 64 + r;

    v8f acc[4] = {};

    const int arow = tid >> 4;              // 0..15
    const int acol = tid & 15;              // 0..15
    const float* Ag = X + (size_t)(tm * 16 + arow) * K + acol;

    for (int k0 = 0; k0 < K; k0 += 16) {
        // ---- stage A panel (16x16 floats, 1 per thread) ----
        Asm[arow * 18 + acol] = Ag[k0];
        // ---- stage B slab (16x512 floats, 8 x b128 per thread, coalesced) ----
#pragma unroll
        for (int j = 0; j < 8; ++j) {
            int f   = tid + j * 256;        // float4 index in 16x128 grid
            int row = f >> 7;
            int c4  = (f & 127) << 2;
            async_copy_b128(&Bsm[row * 520 + c4],
                            W + (size_t)(k0 + row) * WIDTH + c4);
        }
        async_wait_all();
        __syncthreads();

        // ---- 16 WMMAs on the staged chunk ----
#pragma unroll
        for (int s = 0; s < 4; ++s) {
            v2f a = *(const v2f*)&Asm[r * 18 + s * 4 + kk];
#pragma unroll
            for (int t = 0; t < 4; ++t) {
                v2f b;
                b.x = Bsm[(s * 4 + kk)     * 520 + wcol + t * 16];
                b.y = Bsm[(s * 4 + kk + 1) * 520 + wcol + t * 16];
                acc[t] = __builtin_amdgcn_wmma_f32_16x16x4_f32(
                    false, a, false, b, (short)0, acc[t], false, false);
            }
        }
        __syncthreads();
    }

    // ---- store: C/D layout vgpr i -> M = i + 8*(lane>=16), N = lane&15 ----
    const int mrow = tm * 16 + ((lane >> 4) << 3);
#pragma unroll
    for (int t = 0; t < 4; ++t) {
        float* out = H + (size_t)mrow * WIDTH + wave * 64 + t * 16 + r;
#pragma unroll
        for (int i = 0; i < 8; ++i)
            out[(size_t)i * WIDTH] = acc[t][i];
    }
}

// ===================== attention scores a_src/a_dst [N,H] ====================
__global__ void __launch_bounds__(256) att_scores(const float* __restrict__ H,
                                                  const float* __restrict__ att_s,
                                                  const float* __restrict__ att_d,
                                                  float* __restrict__ a_src,
                                                  float* __restrict__ a_dst) {
    int idx = blockIdx.x * 256 + threadIdx.x;
    if (idx >= N_NODES * HEADS) return;
    int n  = idx >> 3;
    int hd = idx & 7;
    const float* hp = H + (size_t)n * WIDTH + hd * HID;
    const float* as = att_s + hd * HID;
    const float* ad = att_d + hd * HID;
    float ss = 0.f, sd = 0.f;
#pragma unroll 8
    for (int c = 0; c < HID; ++c) {
        float hv = hp[c];
        ss += hv * as[c];
        sd += hv * ad[c];
    }
    a_src[idx] = ss;
    a_dst[idx] = sd;
}

// ===================== per-layer init: zero accum + stats ====================
__global__ void __launch_bounds__(256) init_layer(float* __restrict__ accum,
                                                  unsigned* __restrict__ nmax,
                                                  float* __restrict__ nsum) {
    int idx = blockIdx.x * 256 + threadIdx.x;
    if (idx < N_NODES * WIDTH) accum[idx] = 0.f;
    if (idx < N_NODES * HEADS) { nmax[idx] = ENC_NEG_INF; nsum[idx] = 0.f; }
}

// ===================== edge pass 1: leaky-relu logits + segment max ==========
__global__ void __launch_bounds__(256) edge_logits(const int* __restrict__ src,
                                                   const int* __restrict__ dst,
                                                   const float* __restrict__ a_src,
                                                   const float* __restrict__ a_dst,
                                                   float* __restrict__ elog,
                                                   unsigned* __restrict__ nmax) {
    int idx = blockIdx.x * 256 + threadIdx.x;
    if (idx >= N_EDGES * HEADS) return;
    int e  = idx >> 3;
    int hd = idx & 7;
    int s = src[e], d = dst[e];
    float v = a_src[s * HEADS + hd] + a_dst[d * HEADS + hd];
    v = v > 0.f ? v : NEG_SLOPE * v;
    elog[idx] = v;
    atomicMax(&nmax[d * HEADS + hd], encF(v));
}

// ===================== edge pass 2: exp(e - max) + segment sum ===============
__global__ void __launch_bounds__(256) edge_exp(const int* __restrict__ dst,
                                                float* __restrict__ elog,
                                                const unsigned* __restrict__ nmax,
                                                float* __restrict__ nsum) {
    int idx = blockIdx.x * 256 + threadIdx.x;
    if (idx >= N_EDGES * HEADS) return;
    int e  = idx >> 3;
    int hd = idx & 7;
    int d = dst[e];
    float m  = decF(nmax[d * HEADS + hd]);
    float ex = expf(elog[idx] - m);
    elog[idx] = ex;
    atomicAdd(&nsum[d * HEADS + hd], ex);
}

// ===================== edge pass 3: weighted scatter of messages =============
// One wave per (edge, head): 64 channels, 2 per lane.
__global__ void __launch_bounds__(256) edge_aggr(const int* __restrict__ src,
                                                 const int* __restrict__ dst,
                                                 const float* __restrict__ elog,
                                                 const float* __restrict__ nsum,
                                                 const float* __restrict__ H,
                                                 float* __restrict__ accum) {
    int tid  = blockIdx.x * 256 + threadIdx.x;
    int gw   = tid >> 5;                 // wave index == (edge,head) index
    int lane = threadIdx.x & 31;
    if (gw >= N_EDGES * HEADS) return;
    int e  = gw >> 3;
    int hd = gw & 7;
    int s = src[e], d = dst[e];
    float alpha = elog[gw] / (nsum[d * HEADS + hd] + SM_EPS);
    const float* hs = H + (size_t)s * WIDTH + hd * HID;
    float*       od = accum + (size_t)d * WIDTH + hd * HID;
    atomicAdd(od + lane,      hs[lane]      * alpha);
    atomicAdd(od + lane + 32, hs[lane + 32] * alpha);
}

// ===================== bias + LayerNorm + ReLU (in place), wave per node =====
__global__ void __launch_bounds__(256) post_ln(float* __restrict__ H,
                                               const float* __restrict__ bias,
                                               const float* __restrict__ g,
                                               const float* __restrict__ b) {
    int lane = threadIdx.x & 31;
    int node = blockIdx.x * 8 + (threadIdx.x >> 5);   // 50000/8 = 6250 blocks exact
    float* p = H + (size_t)node * WIDTH;
    float v[16];
    float s = 0.f;
#pragma unroll
    for (int i = 0; i < 16; ++i) {
        int c = lane + 32 * i;
        v[i] = p[c] + bias[c];
        s += v[i];
    }
#pragma unroll
    for (int o = 16; o > 0; o >>= 1) s += __shfl_xor(s, o, 32);
    float mu = s * (1.0f / WIDTH);
    float q = 0.f;
#pragma unroll
    for (int i = 0; i < 16; ++i) { float dd = v[i] - mu; q += dd * dd; }
#pragma unroll
    for (int o = 16; o > 0; o >>= 1) q += __shfl_xor(q, o, 32);
    float rs = rsqrtf(q * (1.0f / WIDTH) + LN_EPS);
#pragma unroll
    for (int i = 0; i < 16; ++i) {
        int c = lane + 32 * i;
        float y = (v[i] - mu) * rs * g[c] + b[c];
        p[c] = fmaxf(y, 0.f);
    }
}

// ===================== final FC: out[n] = h[n,:]·fcW + fcb ===================
__global__ void __launch_bounds__(256) final_fc(const float* __restrict__ H,
                                                const float* __restrict__ fcW,
                                                const float* __restrict__ fcb,
                                                float* __restrict__ out) {
    int lane = threadIdx.x & 31;
    int node = blockIdx.x * 8 + (threadIdx.x >> 5);
    const float* p = H + (size_t)node * WIDTH;
    float s = 0.f;
#pragma unroll
    for (int i = 0; i < 16; ++i) {
        int c = lane + 32 * i;
        s += p[c] * fcW[c];
    }
#pragma unroll
    for (int o = 16; o > 0; o >>= 1) s += __shfl_xor(s, o, 32);
    if (lane == 0) out[node] = s + fcb[0];
}

// =============================================================================
extern "C" void kernel_launch(void* const* d_in, const int* in_sizes, int n_in,
                              void* d_out, int out_size, void* d_ws, size_t ws_size,
                              hipStream_t stream) {
    const float* x      = (const float*)d_in[0];
    const int*   edges  = (const int*)d_in[1];     // [2, E] flat
    const float* W1     = (const float*)d_in[2];
    const float* asrc1  = (const float*)d_in[3];
    const float* adst1  = (const float*)d_in[4];
    const float* b1     = (const float*)d_in[5];
    const float* ln1g   = (const float*)d_in[6];
    const float* ln1b   = (const float*)d_in[7];
    const float* W2     = (const float*)d_in[8];
    const float* asrc2  = (const float*)d_in[9];
    const float* adst2  = (const float*)d_in[10];
    const float* b2     = (const float*)d_in[11];
    const float* ln2g   = (const float*)d_in[12];
    const float* ln2b   = (const float*)d_in[13];
    const float* fcW    = (const float*)d_in[14];
    const float* fcb    = (const float*)d_in[15];
    float* out = (float*)d_out;

    const int* src = edges;
    const int* dst = edges + N_EDGES;

    // ---- workspace carve (~237 MB) ----
    float* A    = (float*)d_ws;                          // [N, 512] gemm out
    float* Bacc = A    + (size_t)N_NODES * WIDTH;        // [N, 512] accum / post
    float* aS   = Bacc + (size_t)N_NODES * WIDTH;        // [N, 8]
    float* aD   = aS   + (size_t)N_NODES * HEADS;        // [N, 8]
    unsigned* nmax = (unsigned*)(aD + (size_t)N_NODES * HEADS);  // [N, 8]
    float* nsum = (float*)(nmax + (size_t)N_NODES * HEADS);      // [N, 8]
    float* elog = nsum + (size_t)N_NODES * HEADS;        // [E, 8]

    const int gemmBlocks  = N_NODES / 16;                        // 3125
    const int attBlocks   = (N_NODES * HEADS + 255) / 256;
    const int initBlocks  = (N_NODES * WIDTH + 255) / 256;
    const int edgeBlocks  = (N_EDGES * HEADS + 255) / 256;
    const int aggrBlocks  = (N_EDGES * HEADS * 32) / 256;        // 800000
    const int nodeBlocks  = N_NODES / 8;                         // 6250

    // ---------------- layer 1 ----------------
    gemm_wmma<<<gemmBlocks, 256, 0, stream>>>(x, W1, A, IN_DIM);
    att_scores<<<attBlocks, 256, 0, stream>>>(A, asrc1, adst1, aS, aD);
    init_layer<<<initBlocks, 256, 0, stream>>>(Bacc, nmax, nsum);
    edge_logits<<<edgeBlocks, 256, 0, stream>>>(src, dst, aS, aD, elog, nmax);
    edge_exp<<<edgeBlocks, 256, 0, stream>>>(dst, elog, nmax, nsum);
    edge_aggr<<<aggrBlocks, 256, 0, stream>>>(src, dst, elog, nsum, A, Bacc);
    post_ln<<<nodeBlocks, 256, 0, stream>>>(Bacc, b1, ln1g, ln1b);

    // ---------------- layer 2 ----------------
    gemm_wmma<<<gemmBlocks, 256, 0, stream>>>(Bacc, W2, A, WIDTH);
    att_scores<<<attBlocks, 256, 0, stream>>>(A, asrc2, adst2, aS, aD);
    init_layer<<<initBlocks, 256, 0, stream>>>(Bacc, nmax, nsum);
    edge_logits<<<edgeBlocks, 256, 0, stream>>>(src, dst, aS, aD, elog, nmax);
    edge_exp<<<edgeBlocks, 256, 0, stream>>>(dst, elog, nmax, nsum);
    edge_aggr<<<aggrBlocks, 256, 0, stream>>>(src, dst, elog, nsum, A, Bacc);
    post_ln<<<nodeBlocks, 256, 0, stream>>>(Bacc, b2, ln2g, ln2b);

    // ---------------- head ----------------
    final_fc<<<nodeBlocks, 256, 0, stream>>>(Bacc, fcW, fcb, out);
}